// NormalizedLoss_32581621908072
// MI455X (gfx1250) — compile-verified
//
#include <hip/hip_runtime.h>

typedef float v2f __attribute__((ext_vector_type(2)));
typedef float v8f __attribute__((ext_vector_type(8)));

constexpr int BATCH = 32;
constexpr int NPTS  = 2048;      // N == M == 2048
constexpr float BIGF = 1e10f;

// ---------------------------------------------------------------------------
// Nearest-neighbor pass: for each query point, min/argmin of squared distance
// over all ref points (invalid ref rows pushed out with +1e10 on their norm).
// Block = 256 threads = 8 waves; each wave owns a 16-query column tile.
// Per ref tile of 16: one V_WMMA_F32_16X16X4_F32 computes the 16x16 dot tile.
// Fragment layout (ISA 7.12.2, 32-bit 16x4 A / 4x16 B):
//   lanes 0-15  -> K = {0,1}
//   lanes 16-31 -> K = {2, 0-pad}   (K=3 zero-padded; coords are 3D)
// A fragments are pre-swizzled into LDS as float2[hi][m] so the inner loop is
// a single unconditional ds_load_b64 (no per-iteration selects/branches).
// D: VGPR v, lanes 0-15 -> row v, col lane; lanes 16-31 -> row v+8, col lane-16
// 8 independent min/argmin accumulators (one per D slot) break the serial
// cmp->cndmask dependency chain; slots are merged once after the sweep.
// ---------------------------------------------------------------------------
__global__ void __launch_bounds__(256)
nearest_kernel(const float* __restrict__ qpts, const float* __restrict__ rpts,
               float* __restrict__ out_min, int* __restrict__ out_arg) {
  __shared__ __align__(16) float2 s_af[2 * NPTS];  // [hi][m] A-fragment pairs
  __shared__ __align__(16) float  s_rn[NPTS];      // masked squared norms

  const int tid = threadIdx.x;
  const int b   = blockIdx.y;
  const float* rb = rpts + (size_t)b * NPTS * 3;
  const float* qb = qpts + (size_t)b * NPTS * 3;

  // stage ref points in WMMA-A-fragment layout + masked norms (40KB LDS)
  for (int m = tid; m < NPTS; m += 256) {
    const float c0 = rb[3*m], c1 = rb[3*m+1], c2 = rb[3*m+2];
    s_af[m]        = make_float2(c0, c1);   // hi==0 half: K={0,1}
    s_af[NPTS + m] = make_float2(c2, 0.0f); // hi==1 half: K={2,pad}
    float rn = fmaf(c0, c0, fmaf(c1, c1, c2 * c2));
    if (c0 + c1 + c2 == 0.0f) rn += BIGF;   // exclude invalid ref rows from min
    s_rn[m] = rn;
  }
  __syncthreads();

  const int wave = tid >> 5;
  const int lane = tid & 31;
  const int col  = lane & 15;
  const int hi   = lane >> 4;              // 0: K={0,1}   1: K={2,pad}
  const int n    = blockIdx.x * 128 + wave * 16 + col;

  const float* qp = qb + 3 * n;
  const float q0 = qp[0], q1 = qp[1], q2 = qp[2];
  const float qn = fmaf(q0, q0, fmaf(q1, q1, q2 * q2));

  v2f bfrag;                                // B-matrix: query columns
  bfrag.x = hi ? q2   : q0;
  bfrag.y = hi ? 0.0f : q1;

  const int fbase = hi * NPTS;              // this half-wave's fragment row

  float bv[8];                              // 8 independent min chains
  int   bi[8];
#pragma unroll
  for (int v = 0; v < 8; ++v) { bv[v] = 3.0e38f; bi[v] = 0; }

#pragma unroll 2
  for (int mt = 0; mt < NPTS; mt += 16) {
    const float2 af = s_af[fbase + mt + col];     // one ds_load_b64
    v2f afrag; afrag.x = af.x; afrag.y = af.y;

    v8f acc = {0.f,0.f,0.f,0.f,0.f,0.f,0.f,0.f};
    acc = __builtin_amdgcn_wmma_f32_16x16x4_f32(
        false, afrag, false, bfrag, (short)0, acc, false, false);

    const int mbase = mt + (hi << 3);
    const float4* rn4 = reinterpret_cast<const float4*>(s_rn + mbase);
    const float4 rA = rn4[0], rB = rn4[1];
    const float rv[8] = {rA.x, rA.y, rA.z, rA.w, rB.x, rB.y, rB.z, rB.w};
#pragma unroll
    for (int v = 0; v < 8; ++v) {
      const float dist = fmaf(acc[v], -2.0f, rv[v] + qn);
      // strict < keeps first occurrence: per-slot m indices strictly ascend
      const bool better = dist < bv[v];
      bv[v] = better ? dist : bv[v];
      bi[v] = better ? (mbase + v) : bi[v];
    }
  }

  // merge the 8 slots (branchless, tie-break on smaller m index)
  float best  = bv[0];
  int   bestm = bi[0];
#pragma unroll
  for (int v = 1; v < 8; ++v) {
    const bool take = (bv[v] < best) | ((bv[v] == best) & (bi[v] < bestm));
    best  = take ? bv[v] : best;
    bestm = take ? bi[v] : bestm;
  }

  // merge the two half-wave row partitions (lane L and L+16 share a column)
  const float ob = __shfl_xor(best, 16, 32);
  const int   oa = __shfl_xor(bestm, 16, 32);
  const bool take = (ob < best) | ((ob == best) & (oa < bestm));
  best  = take ? ob : best;
  bestm = take ? oa : bestm;
  if (hi == 0) {
    out_min[(size_t)b * NPTS + n] = best;
    out_arg[(size_t)b * NPTS + n] = bestm;
  }
}

// ---------------------------------------------------------------------------
__global__ void zero_kernel(float* __restrict__ hit_y, float* __restrict__ hit_x) {
  const int i = blockIdx.x * blockDim.x + threadIdx.x;
  if (i < BATCH * NPTS) { hit_y[i] = 0.0f; hit_x[i] = 0.0f; }
}

// hit[b, arg[b,i]] = 1 for every valid query point (races write identical 1.0f)
__global__ void scatter_kernel(const float* __restrict__ qpts,
                               const int* __restrict__ arg,
                               float* __restrict__ hit) {
  const int i = blockIdx.x * blockDim.x + threadIdx.x;
  if (i >= BATCH * NPTS) return;
  const int b = i / NPTS;
  const float* p = qpts + (size_t)3 * i;
  if (p[0] + p[1] + p[2] != 0.0f) hit[(size_t)b * NPTS + arg[i]] = 1.0f;
}

// ---------------------------------------------------------------------------
__device__ float block_reduce_sum(float v, float* red, int tid) {
  red[tid] = v; __syncthreads();
#pragma unroll
  for (int s = 128; s > 0; s >>= 1) {
    if (tid < s) red[tid] += red[tid + s];
    __syncthreads();
  }
  const float r = red[0];
  __syncthreads();
  return r;
}

__global__ void __launch_bounds__(256)
reduce_kernel(const float* __restrict__ x, const float* __restrict__ y,
              const float* __restrict__ x_min, const float* __restrict__ y_min,
              const float* __restrict__ hit_y, const float* __restrict__ hit_x,
              float* __restrict__ bstats) {   // bstats[b] = {cd, cov, qual}
  __shared__ float red[256];
  const int b = blockIdx.x, tid = threadIdx.x;
  const float* xb = x + (size_t)b * NPTS * 3;
  const float* yb = y + (size_t)b * NPTS * 3;
  const float* xm = x_min + (size_t)b * NPTS;
  const float* ym = y_min + (size_t)b * NPTS;
  const float* hy = hit_y + (size_t)b * NPTS;
  const float* hx = hit_x + (size_t)b * NPTS;

  float s_nx = 0, s_xm = 0, s_hx = 0, s_ny = 0, s_ym = 0, s_hy = 0;
  for (int n = tid; n < NPTS; n += 256) {
    float fx = (xb[3*n] + xb[3*n+1] + xb[3*n+2] != 0.0f) ? 1.0f : 0.0f;
    float fy = (yb[3*n] + yb[3*n+1] + yb[3*n+2] != 0.0f) ? 1.0f : 0.0f;
    s_nx += fx; s_xm += xm[n] * fx; s_hx += hx[n];
    s_ny += fy; s_ym += ym[n] * fy; s_hy += hy[n];
  }
  const float nx   = block_reduce_sum(s_nx, red, tid);
  const float xsum = block_reduce_sum(s_xm, red, tid);
  const float hxs  = block_reduce_sum(s_hx, red, tid);
  const float ny   = block_reduce_sum(s_ny, red, tid);
  const float ysum = block_reduce_sum(s_ym, red, tid);
  const float hys  = block_reduce_sum(s_hy, red, tid);
  if (tid == 0) {
    bstats[3*b + 0] = xsum / nx + ysum / ny;  // chamfer
    bstats[3*b + 1] = hys / ny;               // coverage
    bstats[3*b + 2] = hxs / nx;               // quality
  }
}

__global__ void final_kernel(const float* __restrict__ bstats, float* __restrict__ out) {
  if (threadIdx.x == 0 && blockIdx.x == 0) {
    float cd = 0, cov = 0, qual = 0;
    for (int b = 0; b < BATCH; ++b) {
      cd += bstats[3*b]; cov += bstats[3*b+1]; qual += bstats[3*b+2];
    }
    cd /= BATCH; cov /= BATCH; qual /= BATCH;
    out[0] = cd - 1e-4f * cov - 1e-4f * qual;  // val
    out[1] = cd;
    out[2] = cov;
    out[3] = qual;
  }
}

// ---------------------------------------------------------------------------
extern "C" void kernel_launch(void* const* d_in, const int* in_sizes, int n_in,
                              void* d_out, int out_size, void* d_ws, size_t ws_size,
                              hipStream_t stream) {
  const float* x = (const float*)d_in[0];   // [B, N, 3]
  const float* y = (const float*)d_in[1];   // [B, M, 3]
  float* out = (float*)d_out;               // 4 scalars: val, cd, cov, qual

  char* ws = (char*)d_ws;
  const size_t REG = (size_t)BATCH * NPTS * sizeof(float);  // 256 KB
  float* x_min  = (float*)(ws + 0 * REG);
  int*   x_arg  = (int*)  (ws + 1 * REG);
  float* y_min  = (float*)(ws + 2 * REG);
  int*   y_arg  = (int*)  (ws + 3 * REG);
  float* hit_y  = (float*)(ws + 4 * REG);
  float* hit_x  = (float*)(ws + 5 * REG);
  float* bstats = (float*)(ws + 6 * REG);   // 3*B floats

  const dim3 ngrid(NPTS / 128, BATCH);      // 8 waves x 16 queries per block
  nearest_kernel<<<ngrid, 256, 0, stream>>>(x, y, x_min, x_arg);  // per-x nearest y
  nearest_kernel<<<ngrid, 256, 0, stream>>>(y, x, y_min, y_arg);  // per-y nearest x

  const int npt = BATCH * NPTS;
  zero_kernel<<<(npt + 255) / 256, 256, 0, stream>>>(hit_y, hit_x);
  scatter_kernel<<<(npt + 255) / 256, 256, 0, stream>>>(x, x_arg, hit_y);
  scatter_kernel<<<(npt + 255) / 256, 256, 0, stream>>>(y, y_arg, hit_x);

  reduce_kernel<<<BATCH, 256, 0, stream>>>(x, y, x_min, y_min, hit_y, hit_x, bstats);
  final_kernel<<<1, 32, 0, stream>>>(bstats, out);
}